// KPConv_80573586473571
// MI455X (gfx1250) — compile-verified
//
#include <hip/hip_runtime.h>

// KPConv fused kernel for MI455X (gfx1250), f32 WMMA path.
// Stage A: weighted[n,k,c] = sum_h all_w[n,h,k] * x[idx[n,h],c]   (per-query 32x64 GEMMs)
// Stage B: out[n,o]       = sum_{kc} weighted[n,kc] * Wflat[kc,o] (16-query tile GEMM)

typedef float v2f __attribute__((ext_vector_type(2)));
typedef float v8f __attribute__((ext_vector_type(8)));

#define HN        32            // neighbors per query
#define CIN       64
#define COUT      64
#define KPTS      27            // kernel points
#define QT        16            // queries per block (one WMMA M tile)
#define KC        (KPTS * CIN)  // 1728 — flattened reduction length, divisible by 8
#define LW_STRIDE (KC + 4)      // pad to avoid LDS bank conflicts (1732 % 64 == 4)
#define INV_EXT   1.1547005383792515f  // 1 / (RADIUS/(KS-1)*sqrt(3)) = 1/0.8660254

__device__ __forceinline__ float kp_weight(float nx, float ny, float nz,
                                           float kx, float ky, float kz,
                                           float valid) {
  float dx = nx - kx, dy = ny - ky, dz = nz - kz;
  // raw v_sqrt_f32: inputs are finite non-negative, no IEEE fixup tail needed
  float r = __builtin_amdgcn_sqrtf(dx * dx + dy * dy + dz * dz);
  return fmaxf(1.0f - r * INV_EXT, 0.0f) * valid;
}

__global__ __launch_bounds__(128) void kpconv_kernel(
    const float* __restrict__ q_pts,   // [N,3]
    const float* __restrict__ s_pts,   // [M,3]
    const int*   __restrict__ nidx,    // [N,H]
    const float* __restrict__ xg,      // [M,CIN]
    const float* __restrict__ Wf,      // [K,CIN,COUT] flat
    const float* __restrict__ kp,      // [K,3]
    float*       __restrict__ out,     // [N,COUT]
    int n, int m) {
  extern __shared__ char smem[];
  float* nbh  = (float*)smem;                              // [QT][HN][3]
  int*   inds = (int*)(smem + QT * HN * 3 * 4);            // [QT][HN]
  float* lw   = (float*)(smem + QT * HN * 3 * 4 + QT * HN * 4); // [QT][LW_STRIDE]

  const int tid  = threadIdx.x;
  const int lane = tid & 31;
  const int wave = tid >> 5;
  const int q0   = blockIdx.x * QT;
  const int koff = (lane & 16) ? 2 : 0;  // K-slot offset for upper lane half
  const int ccol = lane & 15;            // N column within a 16-wide tile
  const int rb   = (lane & 16) ? 8 : 0;  // C/D row offset for upper lane half

  // ---------------- Phase 0: gather indices + relative positions ----------------
  for (int p = tid; p < QT * HN; p += blockDim.x) {
    int q  = p >> 5, h = p & 31;
    int qg = q0 + q;
    int qc = (qg < n) ? qg : (n - 1);
    int idx    = nidx[qc * HN + h];
    int shadow = (idx >= m);
    int ci     = shadow ? (m - 1) : idx;   // clamped: always a valid gather address
    inds[p] = ci;
    float qx = q_pts[qc * 3 + 0], qy = q_pts[qc * 3 + 1], qz = q_pts[qc * 3 + 2];
    float sx, sy, sz;
    if (shadow) { sx = sy = sz = 1.0e9f; }  // huge distance -> weight clamps to 0
    else { sx = s_pts[ci * 3 + 0]; sy = s_pts[ci * 3 + 1]; sz = s_pts[ci * 3 + 2]; }
    nbh[p * 3 + 0] = sx - qx;
    nbh[p * 3 + 1] = sy - qy;
    nbh[p * 3 + 2] = sz - qz;
  }
  __syncthreads();

  // Per-lane kernel points for k-tile 0 (k = 0..15) and k-tile 1 (k = 16..31).
  int kk0 = ccol;
  int kk1 = 16 + ccol;
  float vld1 = (kk1 < KPTS) ? 1.0f : 0.0f;
  float k0x = kp[kk0 * 3 + 0], k0y = kp[kk0 * 3 + 1], k0z = kp[kk0 * 3 + 2];
  float k1x = 0.0f, k1y = 0.0f, k1z = 0.0f;
  if (kk1 < KPTS) { k1x = kp[kk1 * 3 + 0]; k1y = kp[kk1 * 3 + 1]; k1z = kp[kk1 * 3 + 2]; }

  // ---------------- Phase 1: per-query small GEMMs via WMMA f32 16x16x4 ----------
  for (int qq = 0; qq < QT; qq += 4) {
    int q = qq + wave;

    // A fragments: A[k][h] = all_w[h][k]; lanes 0-15 hold K-slots {0,1}, 16-31 {2,3}.
    v2f a0[8], a1[8];
    int i0[8], i1[8];
#pragma unroll
    for (int hs = 0; hs < 8; ++hs) {
      int hb = hs * 4 + koff;
      const float* nb0 = &nbh[(q * HN + hb) * 3];
      const float* nb1 = nb0 + 3;
      i0[hs] = inds[q * HN + hb];
      i1[hs] = inds[q * HN + hb + 1];
      float n0x = nb0[0], n0y = nb0[1], n0z = nb0[2];
      float n1x = nb1[0], n1y = nb1[1], n1z = nb1[2];
      a0[hs].x = kp_weight(n0x, n0y, n0z, k0x, k0y, k0z, 1.0f);
      a0[hs].y = kp_weight(n1x, n1y, n1z, k0x, k0y, k0z, 1.0f);
      a1[hs].x = kp_weight(n0x, n0y, n0z, k1x, k1y, k1z, vld1);
      a1[hs].y = kp_weight(n1x, n1y, n1z, k1x, k1y, k1z, vld1);
    }

    for (int ct = 0; ct < 4; ++ct) {
      int c = ct * 16 + ccol;
      v8f acc0 = {0.f, 0.f, 0.f, 0.f, 0.f, 0.f, 0.f, 0.f};
      v8f acc1 = {0.f, 0.f, 0.f, 0.f, 0.f, 0.f, 0.f, 0.f};
#pragma unroll
      for (int hs = 0; hs < 8; ++hs) {
        v2f b;
        b.x = xg[i0[hs] * CIN + c];   // L2-resident gather
        b.y = xg[i1[hs] * CIN + c];
        acc0 = __builtin_amdgcn_wmma_f32_16x16x4_f32(false, a0[hs], false, b,
                                                     (short)0, acc0, false, false);
        acc1 = __builtin_amdgcn_wmma_f32_16x16x4_f32(false, a1[hs], false, b,
                                                     (short)0, acc1, false, false);
      }
      // Store D tiles into flattened weighted[q][k*64+c]
      float* lwq = &lw[q * LW_STRIDE];
#pragma unroll
      for (int v = 0; v < 8; ++v) {
        int kr0 = v + rb;            // k in [0,16)
        lwq[kr0 * CIN + c] = acc0[v];
        int kr1 = 16 + v + rb;       // k in [16,32); only k < 27 is real
        if (kr1 < KPTS) lwq[kr1 * CIN + c] = acc1[v];
      }
    }
  }
  __syncthreads();

  // ---------------- Phase 2: out[16q x 64o] = weighted @ Wflat -------------------
  {
    const int o  = wave * 16 + ccol;     // one 16-wide output tile per wave
    const int qa = ccol;                 // A-matrix row = query within tile
    const float* lwq = &lw[qa * LW_STRIDE];
    // two independent accumulator chains to double WMMA-pipe ILP
    v8f acca = {0.f, 0.f, 0.f, 0.f, 0.f, 0.f, 0.f, 0.f};
    v8f accb = {0.f, 0.f, 0.f, 0.f, 0.f, 0.f, 0.f, 0.f};
#pragma unroll 4
    for (int step = 0; step < KC / 8; ++step) {   // 216 iterations, 2 WMMAs each
      int kcA = step * 8 + koff;
      int kcB = kcA + 4;
      v2f aA = *(const v2f*)(&lwq[kcA]);          // ds_load_b64 (8B aligned)
      v2f aB = *(const v2f*)(&lwq[kcB]);
      v2f bA, bB;
      bA.x = Wf[kcA * COUT + o];
      bA.y = Wf[(kcA + 1) * COUT + o];
      bB.x = Wf[kcB * COUT + o];
      bB.y = Wf[(kcB + 1) * COUT + o];
      acca = __builtin_amdgcn_wmma_f32_16x16x4_f32(false, aA, false, bA,
                                                   (short)0, acca, false, false);
      accb = __builtin_amdgcn_wmma_f32_16x16x4_f32(false, aB, false, bB,
                                                   (short)0, accb, false, false);
    }
    v8f acc = acca + accb;
#pragma unroll
    for (int v = 0; v < 8; ++v) {
      int qg = q0 + v + rb;
      if (qg < n) out[qg * COUT + o] = acc[v];
    }
  }
}

extern "C" void kernel_launch(void* const* d_in, const int* in_sizes, int n_in,
                              void* d_out, int out_size, void* d_ws, size_t ws_size,
                              hipStream_t stream) {
  (void)n_in; (void)out_size; (void)d_ws; (void)ws_size;
  const float* q_pts = (const float*)d_in[0];
  const float* s_pts = (const float*)d_in[1];
  const int*   nidx  = (const int*)d_in[2];
  const float* xg    = (const float*)d_in[3];
  const float* Wf    = (const float*)d_in[4];
  const float* kp    = (const float*)d_in[5];
  float* out = (float*)d_out;

  int n = in_sizes[0] / 3;   // 40000 queries
  int m = in_sizes[1] / 3;   // 40000 support points

  size_t shmem = (size_t)(QT * HN * 3 * 4)    // neighbors
               + (size_t)(QT * HN * 4)        // indices
               + (size_t)(QT * LW_STRIDE * 4);// weighted
  hipFuncSetAttribute((const void*)kpconv_kernel,
                      hipFuncAttributeMaxDynamicSharedMemorySize, (int)shmem);

  dim3 grid((n + QT - 1) / QT);
  kpconv_kernel<<<grid, 128, shmem, stream>>>(q_pts, s_pts, nidx, xg, Wf, kp, out, n, m);
}